// DetectionLayer_1717986918798
// MI455X (gfx1250) — compile-verified
//
#include <hip/hip_runtime.h>
#include <math.h>

// Problem constants (must match reference)
#define BATCH    8
#define NROI     2000
#define NCLS     81
#define SORT_N   2048        // NROI padded to power of two for bitonic sort
#define BLOCK    1024        // 32 wave32 waves, one workgroup per batch
#define TILE     256         // ROIs staged per async tile (256*81*4 = 81.0 KB)
#define GROUP    512         // NMS rows per LDS bitmask group
#define NGROUP   (SORT_N / GROUP)     // 4
#define NWORD    (SORT_N / 32)        // 64 u32 words per suppression row
#define MAX_INST 100
#define MIN_CONF 0.7f
#define NMS_THR  0.3f

#define STAGE_WORDS (TILE * NCLS)             // 20736 words per stage buffer
#define KISSUE      6                         // ceil((TILE*NCLS/4)/BLOCK) async ops/thread/tile

// ---- LDS layout (word offsets into dynamic shared) -------------------------
#define OFF_KEY   0                       // f32[2048]  sort keys
#define OFF_IDX   (OFF_KEY  + SORT_N)     // i32[2048]  payload index
#define OFF_BOXP  (OFF_IDX  + SORT_N)     // f32[2048*4] boxes, pre-sort order
#define OFF_CLSP  (OFF_BOXP + 4*SORT_N)   // i32[2048]  class, pre-sort order
#define OFF_KEEP  (OFF_CLSP + SORT_N)     // i32[2048]  final keep flags
#define OFF_U     (OFF_KEEP + SORT_N)     // ---- union region (time-shared) --
#define OFF_BOXS  OFF_U                   // f32[2048*4] boxes, sorted order
#define OFF_CLSS  (OFF_BOXS + 4*SORT_N)   // i32[2048]  class, sorted order
#define OFF_VALW  (OFF_CLSS + SORT_N)     // u32[64]    validity bitset
#define OFF_BCAST (OFF_VALW + NWORD)      // u32[1]     intra-wave broadcast
#define OFF_CNT   (OFF_BCAST + 8)         // i32[128]   per-class counters
#define OFF_SUP   (OFF_CNT + 128)         // u32[512*64] suppression bit rows
#define LDS_WORDS (OFF_SUP + GROUP*NWORD) // 59592 words = 238368 B (<320KB/WGP)
// Phase-A double buffer (2*20736 = 41472 words) overlays the union region
// (43208 words available), both 16B aligned (OFF_U = 16384 words = 64KB).

// ---------------------------------------------------------------------------
// CDNA5 async global->LDS copy (gfx1250), 16B granularity. Tracked by
// ASYNCcnt (per-wave, in-order) -> partial waits enable double buffering.
// ---------------------------------------------------------------------------
__device__ __forceinline__ void async_ld_b128(unsigned lds_byte_off, const float* g) {
    asm volatile("global_load_async_to_lds_b128 %0, %1, off"
                 :
                 : "v"(lds_byte_off), "v"((unsigned long long)(size_t)g)
                 : "memory");
}

template <int N>
__device__ __forceinline__ void wait_async() {
#if __has_builtin(__builtin_amdgcn_s_wait_asynccnt)
    __builtin_amdgcn_s_wait_asynccnt(N);
#else
    asm volatile("s_wait_asynccnt %0" :: "i"(N) : "memory");
#endif
}

// Issue exactly KISSUE async b128 ops per thread (clamped tail => uniform
// per-wave ASYNCcnt increments, so wait_async<KISSUE>() drains the previous
// tile while leaving this tile's copies in flight).
__device__ __forceinline__ void issue_tile(const float* gsrc, unsigned stage_off,
                                           int nvec, int tid) {
#pragma unroll
    for (int it = 0; it < KISSUE; ++it) {
        int q = tid + it * BLOCK;
        if (q >= nvec) q = nvec - 1;        // redundant same-address copy: benign
        async_ld_b128(stage_off + 16u * (unsigned)q, gsrc + 4 * q);
    }
}

extern "C" __global__ __launch_bounds__(BLOCK)
void detection_nms_kernel(const float* __restrict__ rois,     // (B,N,4)
                          const float* __restrict__ probs,    // (B,N,C)
                          const float* __restrict__ deltas,   // (B,N,C,4)
                          float* __restrict__ out)            // (B,100,6)
{
    extern __shared__ float smem[];
    float*    keyS   = smem + OFF_KEY;
    int*      idxS   = (int*)(smem + OFF_IDX);
    float*    boxP   = smem + OFF_BOXP;
    int*      clsP   = (int*)(smem + OFF_CLSP);
    int*      keepS  = (int*)(smem + OFF_KEEP);
    float*    boxS   = smem + OFF_BOXS;
    int*      clsS   = (int*)(smem + OFF_CLSS);
    unsigned* validW = (unsigned*)(smem + OFF_VALW);
    volatile unsigned* bcast = (volatile unsigned*)(smem + OFF_BCAST);
    int*      cnt    = (int*)(smem + OFF_CNT);
    unsigned* supbuf = (unsigned*)(smem + OFF_SUP);
    float*    stage0 = smem + OFF_U;                  // Phase-A double buffer
    float*    stage1 = smem + OFF_U + STAGE_WORDS;

    const int b   = blockIdx.x;
    const int tid = threadIdx.x;

    // ---- pad entries [NROI, SORT_N) ----------------------------------------
    for (int i = NROI + tid; i < SORT_N; i += BLOCK) {
        keyS[i] = -INFINITY;
        idxS[i] = i;
        clsP[i] = -1;
        boxP[4 * i + 0] = 0.f; boxP[4 * i + 1] = 0.f;
        boxP[4 * i + 2] = 0.f; boxP[4 * i + 3] = 0.f;
    }

    // ---- Phase A: argmax + refine; probs double-buffered via async DMA -----
    const size_t probs_base = (size_t)b * NROI * NCLS;
    const unsigned soff0 = (unsigned)(size_t)stage0;
    const unsigned soff1 = (unsigned)(size_t)stage1;
    const int ntiles = (NROI + TILE - 1) / TILE;

    // preload tile 0
    issue_tile(probs + probs_base, soff0, (TILE * NCLS) >> 2, tid);

    for (int t = 0; t < ntiles; ++t) {
        const int r0   = t * TILE;
        const int rows = (NROI - r0 < TILE) ? (NROI - r0) : TILE;
        const float* curstage = (t & 1) ? stage1 : stage0;

        if (t + 1 < ntiles) {
            // issue next tile into the other buffer, then drain only tile t
            const int nrows = (NROI - (r0 + TILE) < TILE) ? (NROI - (r0 + TILE)) : TILE;
            issue_tile(probs + probs_base + (size_t)(r0 + TILE) * NCLS,
                       (t & 1) ? soff0 : soff1, (nrows * NCLS) >> 2, tid);
            wait_async<KISSUE>();        // <=KISSUE outstanding == tile t done
        } else {
            wait_async<0>();
        }
        __syncthreads();

        // speculative prefetch of tile t+2 (gfx1250 global_prefetch)
        if (t + 2 < ntiles) {
            const float* nsrc = probs + probs_base + (size_t)(r0 + 2 * TILE) * NCLS;
            const int nrows2 = (NROI - (r0 + 2 * TILE) < TILE) ? (NROI - (r0 + 2 * TILE)) : TILE;
            int pw = tid * 32;           // every 128 bytes
            if (pw < nrows2 * NCLS) __builtin_prefetch(nsrc + pw, 0, 1);
        }

        // 4 lanes per ROI row: strided class scan + wave32 shuffle reduction.
        const int rpt = tid >> 2;        // row within tile (0..255)
        const int sub = tid & 3;         // sub-scanner (same wave32 wave)
        if (rpt < rows) {
            const int r = r0 + rpt;
            const float* p = curstage + rpt * NCLS;
            float best = -INFINITY;
            int   bc   = 0;
            for (int c = sub; c < NCLS; c += 4) {   // '>' keeps lowest c in subset
                float v = p[c];
                if (v > best) { best = v; bc = c; }
            }
            // butterfly combine: (score desc, class asc) == first-occurrence argmax
#pragma unroll
            for (int d = 1; d < 4; d <<= 1) {
                float so = __shfl_xor(best, d, 32);
                int   co = __shfl_xor(bc,   d, 32);
                if (so > best || (so == best && co < bc)) { best = so; bc = co; }
            }
            if (sub == 0) {
                // class-specific delta: 4 consecutive floats, 16B aligned
                const float4 d4 = *(const float4*)(deltas + ((((size_t)b * NROI + r) * NCLS) + bc) * 4);
                const float4 rb = *(const float4*)(rois + ((size_t)b * NROI + r) * 4);

                float h  = rb.z - rb.x;
                float w_ = rb.w - rb.y;
                float cy = rb.x + 0.5f * h  + (d4.x * 0.1f) * h;
                float cx = rb.y + 0.5f * w_ + (d4.y * 0.1f) * w_;
                h  *= expf(d4.z * 0.2f);
                w_ *= expf(d4.w * 0.2f);
                float y1 = fminf(fmaxf(cy - 0.5f * h,  0.f), 1.f);
                float x1 = fminf(fmaxf(cx - 0.5f * w_, 0.f), 1.f);
                float y2 = fminf(fmaxf(cy + 0.5f * h,  0.f), 1.f);
                float x2 = fminf(fmaxf(cx + 0.5f * w_, 0.f), 1.f);

                const bool valid = (bc > 0) && (best >= MIN_CONF);
                keyS[r] = valid ? best : -INFINITY;
                idxS[r] = r;
                clsP[r] = bc;
                boxP[4 * r + 0] = y1; boxP[4 * r + 1] = x1;
                boxP[4 * r + 2] = y2; boxP[4 * r + 3] = x2;
            }
        }
        __syncthreads();   // readers done before tile t+2 overwrites this buffer
    }

    // ---- Phase B: bitonic sort (score desc, index asc == stable argsort) ---
    for (int k = 2; k <= SORT_N; k <<= 1) {
        for (int j = k >> 1; j > 0; j >>= 1) {
            for (int i = tid; i < SORT_N; i += BLOCK) {
                const int ixj = i ^ j;
                if (ixj > i) {
                    float ka = keyS[i], kb = keyS[ixj];
                    int   ia = idxS[i], ib = idxS[ixj];
                    const bool b_first = (kb > ka) || ((kb == ka) && (ib < ia));
                    if (b_first == ((i & k) == 0)) {
                        keyS[i] = kb; keyS[ixj] = ka;
                        idxS[i] = ib; idxS[ixj] = ia;
                    }
                }
            }
            __syncthreads();
        }
    }

    // ---- gather sorted boxes/classes; build validity bitset ----------------
    for (int i = tid; i < SORT_N; i += BLOCK) {
        const int p = idxS[i];
        boxS[4 * i + 0] = boxP[4 * p + 0];
        boxS[4 * i + 1] = boxP[4 * p + 1];
        boxS[4 * i + 2] = boxP[4 * p + 2];
        boxS[4 * i + 3] = boxP[4 * p + 3];
        clsS[i] = clsP[p];
    }
    if (tid < NWORD) {
        unsigned v = 0u;
        for (int t = 0; t < 32; ++t)
            if (keyS[tid * 32 + t] != -INFINITY) v |= (1u << t);
        validW[tid] = v;
    }
    __syncthreads();

    // ---- Phase C: bitmask NMS ----------------------------------------------
    // removed accumulator in wave-0 registers: lane L owns words L, L+32.
    // Invalid entries start "removed" => they never suppress (keep[i] gating).
    unsigned rem0 = 0u, rem1 = 0u;
    if (tid < 32) { rem0 = ~validW[tid]; rem1 = ~validW[tid + 32]; }

    for (int g = 0; g < NGROUP; ++g) {
        const int rowbase = g * GROUP;

        // (1) all 1024 threads build suppression bit rows for this group
        for (int T = tid; T < GROUP * NWORD; T += BLOCK) {
            const int ri = T >> 6;            // row within group
            const int w  = T & (NWORD - 1);   // 32-column word
            const int i  = rowbase + ri;
            const int j0 = w << 5;
            unsigned bits = 0u;
            if (j0 + 31 > i) {                // word has at least one j > i
                const int    ci = clsS[i];
                const float4 A  = *(const float4*)(boxS + 4 * i);
                const float  areaA = (A.z - A.x) * (A.w - A.y);
                for (int t = 0; t < 32; ++t) {
                    const int j = j0 + t;
                    if (j > i && clsS[j] == ci) {     // class prune (80/81 skip)
                        const float4 Bx = *(const float4*)(boxS + 4 * j);
                        const float iy1 = fmaxf(A.x, Bx.x), ix1 = fmaxf(A.y, Bx.y);
                        const float iy2 = fminf(A.z, Bx.z), ix2 = fminf(A.w, Bx.w);
                        const float inter = fmaxf(iy2 - iy1, 0.f) * fmaxf(ix2 - ix1, 0.f);
                        const float areaB = (Bx.z - Bx.x) * (Bx.w - Bx.y);
                        const float iou   = inter / fmaxf(areaA + areaB - inter, 1e-12f);
                        if (iou > NMS_THR) bits |= (1u << t);
                    }
                }
            }
            supbuf[ri * NWORD + w] = bits;
        }
        __syncthreads();

        // (2) serial greedy scan of this group by one wave32 wave (no barriers)
        if (tid < 32) {
            for (int ri = 0; ri < GROUP; ++ri) {
                const int i  = rowbase + ri;
                const int wi = i >> 5;                       // removed-word index
                if (tid == (wi & 31)) *bcast = (wi < 32) ? rem0 : rem1;
                const unsigned wv = *bcast;                  // same-wave LDS: in order
                if (((wv >> (i & 31)) & 1u) == 0u) {         // entry i survives
                    rem0 |= supbuf[ri * NWORD + tid];
                    rem1 |= supbuf[ri * NWORD + tid + 32];
                }
            }
        }
        __syncthreads();   // group's supbuf free for reuse
    }

    // expand removed bitset into keep flags
    if (tid < 32) {
        for (int s = 0; s < 2; ++s) {
            const unsigned w  = s ? rem1 : rem0;
            const int      wi = s * 32 + tid;
            for (int t = 0; t < 32; ++t)
                keepS[wi * 32 + t] = ((w >> t) & 1u) ? 0 : 1;
        }
    }
    __syncthreads();

    // ---- Phase D: per-class cap (<=100) + compacted emit --------------------
    if (tid == 0) {
        for (int c = 0; c < NCLS; ++c) cnt[c] = 0;
        for (int i = 0; i < SORT_N; ++i) {
            if (keepS[i]) {
                const int c = clsS[i];
                if (++cnt[c] > MAX_INST) keepS[i] = 0;
            }
        }
        float* o = out + (size_t)b * MAX_INST * 6;
        int kout = 0;
        for (int i = 0; i < SORT_N && kout < MAX_INST; ++i) {
            if (keepS[i]) {
                o[kout * 6 + 0] = boxS[4 * i + 0];
                o[kout * 6 + 1] = boxS[4 * i + 1];
                o[kout * 6 + 2] = boxS[4 * i + 2];
                o[kout * 6 + 3] = boxS[4 * i + 3];
                o[kout * 6 + 4] = (float)clsS[i];
                o[kout * 6 + 5] = keyS[i];
                ++kout;
            }
        }
        for (; kout < MAX_INST; ++kout)
            for (int q = 0; q < 6; ++q) o[kout * 6 + q] = 0.f;
    }
}

extern "C" void kernel_launch(void* const* d_in, const int* in_sizes, int n_in,
                              void* d_out, int out_size, void* d_ws, size_t ws_size,
                              hipStream_t stream) {
    (void)in_sizes; (void)n_in; (void)out_size; (void)d_ws; (void)ws_size;
    const float* rois   = (const float*)d_in[0];
    const float* probs  = (const float*)d_in[1];
    const float* deltas = (const float*)d_in[2];
    float* out = (float*)d_out;

    const size_t shmem = (size_t)LDS_WORDS * sizeof(float);   // 238368 B < 320KB

    detection_nms_kernel<<<dim3(BATCH), dim3(BLOCK), shmem, stream>>>(
        rois, probs, deltas, out);
}